// GNNFeatureEncoder_54030688584374
// MI455X (gfx1250) — compile-verified
//
#include <hip/hip_runtime.h>

typedef __attribute__((ext_vector_type(2))) float v2f;
typedef __attribute__((ext_vector_type(8))) float v8f;

#define HDIM 256

// ---------------------------------------------------------------------------
// Degree / normalization coefficient kernels
// ---------------------------------------------------------------------------
__global__ void deg_init_k(unsigned* __restrict__ degU, int n) {
    int i = blockIdx.x * blockDim.x + threadIdx.x;
    if (i < n) degU[i] = 1u;  // self-loop
}

__global__ void deg_edges_k(const int* __restrict__ dst, unsigned* __restrict__ degU, int e) {
    int i = blockIdx.x * blockDim.x + threadIdx.x;
    if (i < e) atomicAdd(&degU[dst[i]], 1u);
}

__global__ void dinv_k(const unsigned* __restrict__ degU, float* __restrict__ dinv, int n) {
    int i = blockIdx.x * blockDim.x + threadIdx.x;
    if (i < n) dinv[i] = rsqrtf((float)degU[i]);  // deg >= 1 always (self-loop)
}

// ---------------------------------------------------------------------------
// WMMA f32 GEMM:  C[N x Hout] = op(A[N x Kin]) @ W[Kin x Hout] (+ bias)
// BN: op = ReLU(BN-affine) fused into the A-fragment load, with the per-
//     channel (scale,shift) pairs staged interleaved in LDS -> one
//     ds_load_b128 per k-step instead of two global_load_b64.
// One wave computes a 16-row x 64-col strip; wave w in block covers cols w*64.
// Requires N % 16 == 0 (N = 50000 = 3125 * 16).
// ---------------------------------------------------------------------------
template <bool BN, bool BIAS>
__global__ __launch_bounds__(128) void gemm_wmma_f32(
    const float* __restrict__ A, const float* __restrict__ W,
    const float* __restrict__ bias, const float* __restrict__ bnScale,
    const float* __restrict__ bnShift, float* __restrict__ C,
    int Kin, int Hout)
{
    __shared__ float ss[2 * HDIM];  // interleaved (scale, shift)

    const int lane = threadIdx.x & 31;
    const int wave = threadIdx.x >> 5;
    const int m    = lane & 15;          // M (A row / C col within tile)
    const int half = lane >> 4;          // 0: K=k,k+1   1: K=k+2,k+3
    const long row0 = (long)blockIdx.x * 16;
    const int  col0 = wave * 64;

    if (BN) {
        for (int c = threadIdx.x; c < Kin; c += blockDim.x) {
            ss[2 * c]     = bnScale[c];
            ss[2 * c + 1] = bnShift[c];
        }
        __syncthreads();
    }

    const float* Arow = A + (row0 + m) * (long)Kin;
    __builtin_prefetch(Arow, 0, 1);      // global_prefetch_b8

    v8f acc0 = {}, acc1 = {}, acc2 = {}, acc3 = {};
#pragma unroll 4
    for (int k = 0; k < Kin; k += 4) {
        const int kk = k + 2 * half;
        v2f a;
        a.x = Arow[kk];
        a.y = Arow[kk + 1];
        if (BN) {
            const float4 q = *(const float4*)&ss[2 * kk];  // ds_load_b128
            a.x = fmaxf(a.x * q.x + q.y, 0.f);
            a.y = fmaxf(a.y * q.z + q.w, 0.f);
        }
        const float* Wr0 = W + (size_t)kk * Hout + col0 + m;
        const float* Wr1 = Wr0 + Hout;
        v2f b0; b0.x = Wr0[0];  b0.y = Wr1[0];
        v2f b1; b1.x = Wr0[16]; b1.y = Wr1[16];
        v2f b2; b2.x = Wr0[32]; b2.y = Wr1[32];
        v2f b3; b3.x = Wr0[48]; b3.y = Wr1[48];
        acc0 = __builtin_amdgcn_wmma_f32_16x16x4_f32(false, a, false, b0, (short)0, acc0, false, false);
        acc1 = __builtin_amdgcn_wmma_f32_16x16x4_f32(false, a, false, b1, (short)0, acc1, false, false);
        acc2 = __builtin_amdgcn_wmma_f32_16x16x4_f32(false, a, false, b2, (short)0, acc2, false, false);
        acc3 = __builtin_amdgcn_wmma_f32_16x16x4_f32(false, a, false, b3, (short)0, acc3, false, false);
    }

    // C layout: VGPR r, lanes<16 -> M=r, lanes>=16 -> M=8+r; N = lane&15.
    float* Cb = C + row0 * Hout;
    const int col = col0 + m;
    float bb0 = 0.f, bb1 = 0.f, bb2 = 0.f, bb3 = 0.f;
    if (BIAS) { bb0 = bias[col]; bb1 = bias[col + 16]; bb2 = bias[col + 32]; bb3 = bias[col + 48]; }
#pragma unroll
    for (int r = 0; r < 8; ++r) {
        const long rr = (long)(r + 8 * half) * Hout;
        Cb[rr + col]      = acc0[r] + bb0;
        Cb[rr + col + 16] = acc1[r] + bb1;
        Cb[rr + col + 32] = acc2[r] + bb2;
        Cb[rr + col + 48] = acc3[r] + bb3;
    }
}

// ---------------------------------------------------------------------------
// agg[i,c] = hW[i,c] * dinv[i]^2 (self-loop term) + b[c]
// ---------------------------------------------------------------------------
__global__ void selfloop_init_k(const float* __restrict__ hW, const float* __restrict__ dinv,
                                const float* __restrict__ b, float* __restrict__ agg, int total) {
    int i = blockIdx.x * blockDim.x + threadIdx.x;
    if (i >= total) return;
    const int row = i >> 8;
    const int c   = i & 255;
    const float di = dinv[row];
    agg[i] = hW[i] * di * di + b[c];
}

// ---------------------------------------------------------------------------
// One wave per edge: gather hW[src] (float4), scale by dinv[s]*dinv[d],
// scatter via hardware global_atomic_add_f32 into agg[dst].
// Both hW and agg are L2-resident (51 MB each vs 192 MB L2).
// ---------------------------------------------------------------------------
__global__ __launch_bounds__(256) void edge_scatter_k(
    const float* __restrict__ hW, const int* __restrict__ src, const int* __restrict__ dst,
    const float* __restrict__ dinv, float* __restrict__ agg, int e)
{
    const int w = (int)((blockIdx.x * (unsigned)blockDim.x + threadIdx.x) >> 5);
    if (w >= e) return;
    const int lane = threadIdx.x & 31;
    const int s = src[w];
    const int d = dst[w];
    const float c = dinv[s] * dinv[d];

    const float4* rp = (const float4*)(hW + (size_t)s * HDIM);
    const float4 lo = rp[lane];        // cols [lane*4 .. lane*4+3]
    const float4 hi = rp[lane + 32];   // cols [128+lane*4 .. ]
    float* base = agg + (size_t)d * HDIM;
    const int c0 = lane * 4;
    unsafeAtomicAdd(base + c0 + 0,       lo.x * c);
    unsafeAtomicAdd(base + c0 + 1,       lo.y * c);
    unsafeAtomicAdd(base + c0 + 2,       lo.z * c);
    unsafeAtomicAdd(base + c0 + 3,       lo.w * c);
    unsafeAtomicAdd(base + 128 + c0 + 0, hi.x * c);
    unsafeAtomicAdd(base + 128 + c0 + 1, hi.y * c);
    unsafeAtomicAdd(base + 128 + c0 + 2, hi.z * c);
    unsafeAtomicAdd(base + 128 + c0 + 3, hi.w * c);
}

// ---------------------------------------------------------------------------
// BatchNorm statistics: sums[0..255] = column sums, sums[256..511] = sum of sq.
// ---------------------------------------------------------------------------
__global__ void zero_sums_k(float* __restrict__ sums) {
    int i = blockIdx.x * blockDim.x + threadIdx.x;
    if (i < 512) sums[i] = 0.f;
}

__global__ __launch_bounds__(256) void bn_stats_k(const float* __restrict__ h,
                                                  float* __restrict__ sums, int n) {
    const int ch = threadIdx.x;  // 256 channels
    float s = 0.f, s2 = 0.f;
    for (int r = blockIdx.x; r < n; r += gridDim.x) {
        const float v = h[(size_t)r * HDIM + ch];
        s += v;
        s2 += v * v;
    }
    unsafeAtomicAdd(&sums[ch], s);
    unsafeAtomicAdd(&sums[HDIM + ch], s2);
}

__global__ void bn_finalize_k(const float* __restrict__ sums, const float* __restrict__ gamma,
                              const float* __restrict__ beta, float* __restrict__ scale,
                              float* __restrict__ shift, float invN) {
    const int c = threadIdx.x;
    const float mean = sums[c] * invN;
    const float var  = sums[HDIM + c] * invN - mean * mean;  // biased var (matches jnp.var)
    const float sc = gamma[c] * rsqrtf(var + 1e-5f);
    scale[c] = sc;
    shift[c] = beta[c] - mean * sc;
}

// ---------------------------------------------------------------------------
extern "C" void kernel_launch(void* const* d_in, const int* in_sizes, int n_in,
                              void* d_out, int out_size, void* d_ws, size_t ws_size,
                              hipStream_t stream) {
    const float* x     = (const float*)d_in[0];
    const int*   ei    = (const int*)d_in[1];
    const float* W0    = (const float*)d_in[2];
    const float* b0    = (const float*)d_in[3];
    const float* g0    = (const float*)d_in[4];
    const float* be0   = (const float*)d_in[5];
    const float* W1    = (const float*)d_in[6];
    const float* b1    = (const float*)d_in[7];
    const float* g1    = (const float*)d_in[8];
    const float* be1   = (const float*)d_in[9];
    const float* W_mu  = (const float*)d_in[10];
    const float* b_mu  = (const float*)d_in[11];
    const float* W_lv  = (const float*)d_in[12];
    const float* b_lv  = (const float*)d_in[13];
    float* out = (float*)d_out;

    const int N = in_sizes[0] / HDIM;   // 50000 (divisible by 16)
    const int E = in_sizes[1] / 2;      // 800000
    const int L = 128;
    const int* src = ei;
    const int* dst = ei + E;

    // Workspace layout (~103 MB)
    float*    bufA = (float*)d_ws;                 // N*256  (aggregation buffer)
    float*    bufB = bufA + (size_t)N * HDIM;      // N*256  (hW buffer)
    float*    dinv = bufB + (size_t)N * HDIM;      // N
    unsigned* degU = (unsigned*)(dinv + N);        // N
    float*    sums = (float*)(degU + N);           // 512
    float*    scal = sums + 512;                   // 256
    float*    shft = scal + HDIM;                  // 256

    const int T = 256;
    const dim3 gN((N + T - 1) / T), gE((E + T - 1) / T);
    const dim3 gElem(((size_t)N * HDIM + T - 1) / T);
    const dim3 gEdgeWaves((unsigned)(((size_t)E * 32 + T - 1) / T));
    const dim3 gGemm(N / 16);
    const float invN = 1.0f / (float)N;

    // --- degrees / symmetric normalization coefficients ---
    deg_init_k<<<gN, T, 0, stream>>>(degU, N);
    deg_edges_k<<<gE, T, 0, stream>>>(dst, degU, E);
    dinv_k<<<gN, T, 0, stream>>>(degU, dinv, N);

    // --- Layer 0: hW0 = x @ W0 ; agg0 = scatter + b0 ; BN stats ---
    gemm_wmma_f32<false, false><<<gGemm, 128, 0, stream>>>(
        x, W0, nullptr, nullptr, nullptr, bufB, HDIM, HDIM);
    selfloop_init_k<<<gElem, T, 0, stream>>>(bufB, dinv, b0, bufA, N * HDIM);
    edge_scatter_k<<<gEdgeWaves, T, 0, stream>>>(bufB, src, dst, dinv, bufA, E);
    zero_sums_k<<<2, T, 0, stream>>>(sums);
    bn_stats_k<<<400, T, 0, stream>>>(bufA, sums, N);
    bn_finalize_k<<<1, T, 0, stream>>>(sums, g0, be0, scal, shft, invN);

    // --- Layer 1: hW1 = relu(bn(agg0)) @ W1 (BN+ReLU fused into A load) ---
    gemm_wmma_f32<true, false><<<gGemm, 128, 0, stream>>>(
        bufA, W1, nullptr, scal, shft, bufB, HDIM, HDIM);
    selfloop_init_k<<<gElem, T, 0, stream>>>(bufB, dinv, b1, bufA, N * HDIM);
    edge_scatter_k<<<gEdgeWaves, T, 0, stream>>>(bufB, src, dst, dinv, bufA, E);
    zero_sums_k<<<2, T, 0, stream>>>(sums);
    bn_stats_k<<<400, T, 0, stream>>>(bufA, sums, N);
    bn_finalize_k<<<1, T, 0, stream>>>(sums, g1, be1, scal, shft, invN);

    // --- Heads: mu = relu(bn(agg1)) @ W_mu + b_mu ; logvar likewise ---
    gemm_wmma_f32<true, true><<<gGemm, 64, 0, stream>>>(
        bufA, W_mu, b_mu, scal, shft, out, HDIM, L);
    gemm_wmma_f32<true, true><<<gGemm, 64, 0, stream>>>(
        bufA, W_lv, b_lv, scal, shft, out + (size_t)N * L, HDIM, L);
}